// GATMultiLabel_13915694039844
// MI455X (gfx1250) — compile-verified
//
#include <hip/hip_runtime.h>
#include <hip/hip_bf16.h>
#include <math.h>

typedef float v2f __attribute__((ext_vector_type(2)));
typedef float v8f __attribute__((ext_vector_type(8)));

#define IN_DIM 512
#define HID    64
#define HEADS  8
#define F1     (HEADS * HID)   // 512
#define OUT64  64
#define NEG_SLOPE 0.2f
#define KC     32              // K-slab staged in LDS per iteration
#define KCP    36              // padded LDS row stride (float4-aligned, bank-spread)

// ---------- order-preserving float<->uint mapping for atomic max ----------
__device__ __forceinline__ unsigned fenc(float f) {
    unsigned u = __float_as_uint(f);
    return (u & 0x80000000u) ? ~u : (u | 0x80000000u);
}
__device__ __forceinline__ float fdec(unsigned u) {
    u = (u & 0x80000000u) ? (u & 0x7FFFFFFFu) : ~u;
    return __uint_as_float(u);
}

// =====================================================================
// WMMA fp32 GEMM:  C[M,Ncols] = A[M,K] @ W[Ncols,K]^T + bias[Ncols]
// 128-thread block = 4 waves; all share one M-tile (A slab staged in LDS
// once per block), each wave owns one 16x16 N-tile. K consumed in slabs
// of 32 -> unrolled clause of 8 V_WMMA_F32_16X16X4_F32 per slab.
// EXEC stays all-ones through every WMMA (loads clamped, stores guarded).
// =====================================================================
__global__ __launch_bounds__(128)
void wmma_gemm_bias_kernel(const float* __restrict__ A,
                           const float* __restrict__ W,
                           const float* __restrict__ bias,
                           float* __restrict__ C,
                           int M, int Ncols, int K) {
    __shared__ float shA[16 * KCP];

    const int lane  = threadIdx.x & 31;
    const int wave  = threadIdx.x >> 5;
    const int tileM = blockIdx.x;
    const int tileN = blockIdx.y * 4 + wave;

    const int n   = tileN * 16 + (lane & 15);
    const int ncl = n < Ncols ? n : Ncols - 1;      // clamp, keep EXEC full
    const int kb  = (lane >> 4) << 1;               // this lane's K pair: 0 or 2
    const int ar  = (lane & 15) * KCP + kb;         // LDS base for A fragment

    // cooperative A-slab loader coords: 16 rows x 32 floats, float4/thread
    const int lrow = threadIdx.x >> 3;              // 0..15
    const int lk   = (threadIdx.x & 7) << 2;        // 0,4,...,28
    int grow = tileM * 16 + lrow;
    grow = grow < M ? grow : M - 1;
    const float* __restrict__ aRow = A + (size_t)grow * K + lk;
    const float* __restrict__ brow = W + (size_t)ncl * K + kb;

    v8f acc = {};
    for (int k0 = 0; k0 < K; k0 += KC) {
        // stage A slab: global -> LDS (one float4 per thread)
        float4 av = *(const float4*)(aRow + k0);
        *(float4*)(&shA[lrow * KCP + lk]) = av;
        __syncthreads();
#pragma unroll
        for (int kk = 0; kk < KC; kk += 4) {
            v2f a, b;
            a.x = shA[ar + kk];
            a.y = shA[ar + kk + 1];
            b.x = brow[k0 + kk];
            b.y = brow[k0 + kk + 1];
            acc = __builtin_amdgcn_wmma_f32_16x16x4_f32(
                /*neg_a=*/false, a, /*neg_b=*/false, b,
                /*c_mod=*/(short)0, acc, /*reuse_a=*/false, /*reuse_b=*/false);
        }
        __syncthreads();
    }

    const bool  colOK = (n < Ncols);
    const float bv    = colOK ? bias[n] : 0.0f;
    const int   mrow  = tileM * 16 + ((lane >> 4) << 3);
#pragma unroll
    for (int v = 0; v < 8; ++v) {
        int row = mrow + v;
        if (colOK && row < M)
            C[(size_t)row * Ncols + n] = acc[v] + bv;
    }
}

// =====================================================================
// Workspace init: zero accumulators & segment sums, reset max keys.
// =====================================================================
__global__ void init_ws_kernel(float* __restrict__ acc, long long nAcc,
                               float* __restrict__ ssum,
                               unsigned* __restrict__ maxk, long long nSeg) {
    long long i = (long long)blockIdx.x * blockDim.x + threadIdx.x;
    if (i < nAcc) acc[i] = 0.0f;
    if (i < nSeg) { ssum[i] = 0.0f; maxk[i] = 0u; }   // 0u == -inf encoded
}

// =====================================================================
// Pass 1: attention logits. One wave per (edge, head).
// e[w] = sum_c att[h][c] * leaky_relu(XL[src][h,c] + XR[dst][h,c])
// Also atomic-max the encoded logit into maxk[dst*H + h].
// =====================================================================
__global__ __launch_bounds__(256)
void edge_logits_kernel(const float* __restrict__ XL,
                        const float* __restrict__ XR,
                        const float* __restrict__ att,
                        const int* __restrict__ srcA,
                        const int* __restrict__ dstA,
                        float* __restrict__ ebuf,
                        unsigned* __restrict__ maxk,
                        int E, int Etot, int H, int Cdim, int F) {
    int w    = (int)(((long long)blockIdx.x * blockDim.x + threadIdx.x) >> 5);
    int lane = threadIdx.x & 31;
    if (w >= Etot * H) return;                  // uniform per wave
    int e = w / H, h = w - e * H;
    int s = (e < E) ? srcA[e] : (e - E);        // self-loops appended
    int d = (e < E) ? dstA[e] : (e - E);

    const float* xlr = XL + (size_t)s * F + h * Cdim;
    const float* xrr = XR + (size_t)d * F + h * Cdim;
    __builtin_prefetch(xlr + lane, 0, 0);       // global_prefetch_b8: start the
    __builtin_prefetch(xrr + lane, 0, 0);       // random-row gathers early
    const float* ar  = att + h * Cdim;

    float dot = 0.0f;
    for (int c = lane; c < Cdim; c += 32) {
        float v = xlr[c] + xrr[c];
        v = (v > 0.0f) ? v : NEG_SLOPE * v;
        dot += ar[c] * v;
    }
#pragma unroll
    for (int off = 16; off > 0; off >>= 1)
        dot += __shfl_xor(dot, off, 32);

    if (lane == 0) {
        ebuf[w] = dot;
        atomicMax(&maxk[(size_t)d * H + h], fenc(dot));
    }
}

// =====================================================================
// Pass 2: a = exp(e - max[dst]); segment-sum via atomicAdd. In-place ebuf.
// One thread per (edge, head).
// =====================================================================
__global__ __launch_bounds__(256)
void edge_exp_kernel(const int* __restrict__ dstA,
                     float* __restrict__ ebuf,
                     const unsigned* __restrict__ maxk,
                     float* __restrict__ ssum,
                     int E, int Etot, int H) {
    long long i = (long long)blockIdx.x * blockDim.x + threadIdx.x;
    if (i >= (long long)Etot * H) return;
    int e = (int)(i / H), h = (int)(i - (long long)e * H);
    int d = (e < E) ? dstA[e] : (e - E);
    float m = fdec(maxk[(size_t)d * H + h]);
    float a = __expf(ebuf[i] - m);
    ebuf[i] = a;
    atomicAdd(&ssum[(size_t)d * H + h], a);
}

// =====================================================================
// Pass 3: alpha = a / (s[dst]+eps); acc[dst][h,c] += XL[src][h,c] * alpha.
// One wave per (edge, head); lanes stride channels (coalesced 256B bursts,
// L2-resident tables -> gathers hit the 192MB L2).
// =====================================================================
__global__ __launch_bounds__(256)
void edge_scatter_kernel(const float* __restrict__ XL,
                         const int* __restrict__ srcA,
                         const int* __restrict__ dstA,
                         const float* __restrict__ ebuf,
                         const float* __restrict__ ssum,
                         float* __restrict__ acc,
                         int E, int Etot, int H, int Cdim, int F) {
    int w    = (int)(((long long)blockIdx.x * blockDim.x + threadIdx.x) >> 5);
    int lane = threadIdx.x & 31;
    if (w >= Etot * H) return;
    int e = w / H, h = w - e * H;
    int s = (e < E) ? srcA[e] : (e - E);
    int d = (e < E) ? dstA[e] : (e - E);

    const float* xlr = XL + (size_t)s * F + h * Cdim;
    __builtin_prefetch(xlr + lane, 0, 0);       // prefetch the gathered row
    float alpha = ebuf[w] / (ssum[(size_t)d * H + h] + 1e-16f);
    float*       out = acc + (size_t)d * F + h * Cdim;
    for (int c = lane; c < Cdim; c += 32)
        atomicAdd(&out[c], xlr[c] * alpha);
}

// =====================================================================
// out = relu(acc + bias[col]) in place.
// =====================================================================
__global__ __launch_bounds__(256)
void bias_relu_kernel(float* __restrict__ acc, const float* __restrict__ bias,
                      long long total, int F) {
    long long i = (long long)blockIdx.x * blockDim.x + threadIdx.x;
    if (i >= total) return;
    float v = acc[i] + bias[(int)(i % F)];
    acc[i] = (v > 0.0f) ? v : 0.0f;
}

// =====================================================================
extern "C" void kernel_launch(void* const* d_in, const int* in_sizes, int n_in,
                              void* d_out, int out_size, void* d_ws, size_t ws_size,
                              hipStream_t stream) {
    const float* x     = (const float*)d_in[0];
    const int*   eidx  = (const int*)  d_in[1];
    const float* W1l   = (const float*)d_in[2];
    const float* b1l   = (const float*)d_in[3];
    const float* W1r   = (const float*)d_in[4];
    const float* b1r   = (const float*)d_in[5];
    const float* att1  = (const float*)d_in[6];
    const float* bias1 = (const float*)d_in[7];
    const float* W2l   = (const float*)d_in[8];
    const float* b2l   = (const float*)d_in[9];
    const float* W2r   = (const float*)d_in[10];
    const float* b2r   = (const float*)d_in[11];
    const float* att2  = (const float*)d_in[12];
    const float* bias2 = (const float*)d_in[13];
    const float* fc_w  = (const float*)d_in[14];
    const float* fc_b  = (const float*)d_in[15];
    float*       out   = (float*)d_out;

    const int NN      = in_sizes[0] / IN_DIM;          // 10000 nodes
    const int E       = in_sizes[1] / 2;               // 160000 edges
    const int ETOT    = E + NN;                        // + self loops
    const int OUT_DIM = in_sizes[14] / OUT64;          // 5000
    const int* srcA = eidx;                            // edge_index[0]
    const int* dstA = eidx + E;                        // edge_index[1]

    // ---- carve workspace ----
    float* ws = (float*)d_ws;
    size_t o = 0;
    float* XL1  = ws + o; o += (size_t)NN * F1;
    float* XR1  = ws + o; o += (size_t)NN * F1;
    float* ACC1 = ws + o; o += (size_t)NN * F1;        // becomes h1
    float* XL2  = ws + o; o += (size_t)NN * OUT64;
    float* XR2  = ws + o; o += (size_t)NN * OUT64;
    float* ACC2 = ws + o; o += (size_t)NN * OUT64;     // becomes h2
    float* EBUF = ws + o; o += (size_t)ETOT * HEADS;   // reused by layer 2
    float* SSUM = ws + o; o += (size_t)NN * HEADS;
    unsigned* MAXK = (unsigned*)(ws + o); o += (size_t)NN * HEADS;
    (void)ws_size; (void)n_in; (void)out_size;

    const dim3 blk(128);

    // ===== Layer 1: node transforms via WMMA, M=NN, N=512, K=512 =====
    {
        dim3 grid((NN + 15) / 16, (F1 + 63) / 64);
        wmma_gemm_bias_kernel<<<grid, blk, 0, stream>>>(x, W1l, b1l, XL1, NN, F1, IN_DIM);
        wmma_gemm_bias_kernel<<<grid, blk, 0, stream>>>(x, W1r, b1r, XR1, NN, F1, IN_DIM);
    }
    // ===== init layer-1 accumulators =====
    {
        long long nAcc = (long long)NN * F1, nSeg = (long long)NN * HEADS;
        long long nmax = nAcc > nSeg ? nAcc : nSeg;
        init_ws_kernel<<<(unsigned)((nmax + 255) / 256), 256, 0, stream>>>(ACC1, nAcc, SSUM, MAXK, nSeg);
    }
    // ===== layer-1 edge passes (H=8, C=64) =====
    {
        long long waves = (long long)ETOT * HEADS;
        unsigned gLog = (unsigned)((waves * 32 + 255) / 256);
        unsigned gThr = (unsigned)((waves + 255) / 256);
        edge_logits_kernel<<<gLog, 256, 0, stream>>>(XL1, XR1, att1, srcA, dstA, EBUF, MAXK,
                                                     E, ETOT, HEADS, HID, F1);
        edge_exp_kernel<<<gThr, 256, 0, stream>>>(dstA, EBUF, MAXK, SSUM, E, ETOT, HEADS);
        edge_scatter_kernel<<<gLog, 256, 0, stream>>>(XL1, srcA, dstA, EBUF, SSUM, ACC1,
                                                      E, ETOT, HEADS, HID, F1);
        long long tot = (long long)NN * F1;
        bias_relu_kernel<<<(unsigned)((tot + 255) / 256), 256, 0, stream>>>(ACC1, bias1, tot, F1);
    }
    // ===== Layer 2: node transforms via WMMA, M=NN, N=64, K=512 =====
    {
        dim3 grid((NN + 15) / 16, (OUT64 + 63) / 64);
        wmma_gemm_bias_kernel<<<grid, blk, 0, stream>>>(ACC1, W2l, b2l, XL2, NN, OUT64, F1);
        wmma_gemm_bias_kernel<<<grid, blk, 0, stream>>>(ACC1, W2r, b2r, XR2, NN, OUT64, F1);
    }
    // ===== init layer-2 accumulators =====
    {
        long long nAcc = (long long)NN * OUT64, nSeg = NN;
        init_ws_kernel<<<(unsigned)((nAcc + 255) / 256), 256, 0, stream>>>(ACC2, nAcc, SSUM, MAXK, nSeg);
    }
    // ===== layer-2 edge passes (H=1, C=64) =====
    {
        long long waves = ETOT;
        unsigned gLog = (unsigned)((waves * 32 + 255) / 256);
        unsigned gThr = (unsigned)((waves + 255) / 256);
        edge_logits_kernel<<<gLog, 256, 0, stream>>>(XL2, XR2, att2, srcA, dstA, EBUF, MAXK,
                                                     E, ETOT, 1, OUT64, OUT64);
        edge_exp_kernel<<<gThr, 256, 0, stream>>>(dstA, EBUF, MAXK, SSUM, E, ETOT, 1);
        edge_scatter_kernel<<<gLog, 256, 0, stream>>>(XL2, srcA, dstA, EBUF, SSUM, ACC2,
                                                      E, ETOT, 1, OUT64, OUT64);
        long long tot = (long long)NN * OUT64;
        bias_relu_kernel<<<(unsigned)((tot + 255) / 256), 256, 0, stream>>>(ACC2, bias2, tot, OUT64);
    }
    // ===== Final FC via WMMA: M=NN, N=OUT_DIM(5000), K=64 =====
    {
        dim3 grid((NN + 15) / 16, (OUT_DIM + 63) / 64);
        wmma_gemm_bias_kernel<<<grid, blk, 0, stream>>>(ACC2, fc_w, fc_b, out, NN, OUT_DIM, OUT64);
    }
}